// TemporalGraphConv_63771674411185
// MI455X (gfx1250) — compile-verified
//
#include <hip/hip_runtime.h>
#include <hip/hip_bf16.h>
#include <math.h>

// ---------------------------------------------------------------------------
// Problem: out[b,c,n,m] = sum_t x[b,c,n,t] * adp[t,m]
//   x: (64,64,2048,12) fp32, adp: 12x12 built from tiny tensors + softmax.
// HBM-bound (AI ~= 3 flop/byte): floor ~35us at 23.3 TB/s. WMMA f32 16x16x4
// provides the compute essentially for free; all effort goes to streaming.
// ---------------------------------------------------------------------------

typedef __attribute__((ext_vector_type(2))) float v2f;
typedef __attribute__((ext_vector_type(8))) float v8f;

#define T_DIM 12
#define D_DIM 10

// ---------------------------------------------------------------------------
// Phase 1: build adp (12x12, relu + row softmax) and pre-pack it into the
// per-lane B-operand layout of V_WMMA_F32_16X16X4_F32:
//   blay[j][lane] = { adp[4j + 2*(lane>>4) + 0][lane&15],
//                     adp[4j + 2*(lane>>4) + 1][lane&15] }   for j = 0..2
// Zero-padded for column index >= 12 so WMMA columns 12..15 are inert.
// ---------------------------------------------------------------------------
__global__ __launch_bounds__(256) void adp_build_kernel(
    const float* __restrict__ timevec1,   // (12,10)
    const float* __restrict__ timevec2,   // (12,10)
    const float* __restrict__ nodevec,    // (288,10)
    const float* __restrict__ kten,       // (10,10,10)
    const int*   __restrict__ time_ind_p, // scalar
    float* __restrict__ blay)             // 3*32 float2 = 192 floats in d_ws
{
  __shared__ float adp1[D_DIM][D_DIM];
  __shared__ float adp2[T_DIM][D_DIM];
  __shared__ float adp [T_DIM][T_DIM];
  const int tid = threadIdx.x;
  const int ti  = time_ind_p[0];

  // adp1[e][f] = sum_d nodevec[ti,d] * k[d,e,f]
  if (tid < D_DIM * D_DIM) {
    const int e = tid / D_DIM, f = tid % D_DIM;
    float s = 0.f;
    #pragma unroll
    for (int d = 0; d < D_DIM; ++d)
      s += nodevec[ti * D_DIM + d] * kten[(d * D_DIM + e) * D_DIM + f];
    adp1[e][f] = s;
  }
  __syncthreads();

  // adp2[b][f] = sum_e timevec1[b,e] * adp1[e,f]
  if (tid < T_DIM * D_DIM) {
    const int b = tid / D_DIM, f = tid % D_DIM;
    float s = 0.f;
    #pragma unroll
    for (int e = 0; e < D_DIM; ++e)
      s += timevec1[b * D_DIM + e] * adp1[e][f];
    adp2[b][f] = s;
  }
  __syncthreads();

  // adp3[b][c] = sum_f timevec2[c,f] * adp2[b,f]; relu
  if (tid < T_DIM * T_DIM) {
    const int b = tid / T_DIM, c = tid % T_DIM;
    float s = 0.f;
    #pragma unroll
    for (int f = 0; f < D_DIM; ++f)
      s += timevec2[c * D_DIM + f] * adp2[b][f];
    adp[b][c] = fmaxf(s, 0.f);
  }
  __syncthreads();

  // Row softmax over the last dim (12 rows, one thread each).
  if (tid < T_DIM) {
    float mx = -1e30f;
    #pragma unroll
    for (int c = 0; c < T_DIM; ++c) mx = fmaxf(mx, adp[tid][c]);
    float sum = 0.f;
    #pragma unroll
    for (int c = 0; c < T_DIM; ++c) {
      const float e = __expf(adp[tid][c] - mx);
      adp[tid][c] = e;
      sum += e;
    }
    const float inv = 1.f / sum;
    #pragma unroll
    for (int c = 0; c < T_DIM; ++c) adp[tid][c] *= inv;
  }
  __syncthreads();

  // Pack into WMMA B-operand lane layout (3 K-groups of 4, K = 0..11).
  if (tid < 3 * 32) {
    const int j = tid / 32, lane = tid % 32;
    const int h = lane >> 4, n = lane & 15;
    const int t0 = 4 * j + 2 * h;
    const float v0 = (n < T_DIM) ? adp[t0 + 0][n] : 0.f;
    const float v1 = (n < T_DIM) ? adp[t0 + 1][n] : 0.f;
    blay[2 * tid + 0] = v0;
    blay[2 * tid + 1] = v1;
  }
}

// ---------------------------------------------------------------------------
// Phase 2: streaming GEMM (R x 12) * (12 x 12) with one 16-row tile per wave.
// A 16-row tile of x (and of out) is 768 contiguous bytes (T=12 innermost).
// 3x v_wmma_f32_16x16x4_f32 per tile; NT loads/stores. The D tile is staged
// through LDS directly in OUTPUT MEMORY LAYOUT (compact 12-wide rows), so
// readback/store indexing is pure lane*2 + i*64 (no div/mod by 12), and the
// half-wave write banks are disjoint (8*12 = 96 ≡ 32 mod 64 banks).
// ---------------------------------------------------------------------------
__global__ __launch_bounds__(256) void temporal_mix_kernel(
    const float* __restrict__ x,     // (R, 12) row-major
    const float* __restrict__ blay,  // packed B operands, 192 floats
    float* __restrict__ out,         // (R, 12) row-major
    long rows)
{
  const int lane = threadIdx.x & 31;
  const int wave = threadIdx.x >> 5;
  const int half = lane >> 4;   // 0: K pair {0,1}; 1: K pair {2,3}
  const int mrow = lane & 15;   // A-row / D-column index

  const long tile     = (long)blockIdx.x * 8 + wave;
  const long row_base = tile << 4;
  if (row_base + 16 > rows) return;  // wave-uniform; never taken for this shape

  // B operands (tiny, L2-hot, identical for every wave).
  const v2f b0 = *(const v2f*)(blay + 2 * (0 * 32 + lane));
  const v2f b1 = *(const v2f*)(blay + 2 * (1 * 32 + lane));
  const v2f b2 = *(const v2f*)(blay + 2 * (2 * 32 + lane));

  // A operands: lane l holds x[row_base + l%16][4j + 2*(l>=16) + {0,1}].
  const float* xrow = x + (row_base + mrow) * T_DIM + 2 * half;
  const v2f a0 = __builtin_nontemporal_load((const v2f*)(xrow + 0));
  const v2f a1 = __builtin_nontemporal_load((const v2f*)(xrow + 4));
  const v2f a2 = __builtin_nontemporal_load((const v2f*)(xrow + 8));

  // D = A*B + C chained over K = 0..11 (K=12..15 implicitly zero).
  v8f acc = {};
  acc = __builtin_amdgcn_wmma_f32_16x16x4_f32(false, a0, false, b0,
                                              (short)0, acc, false, false);
  acc = __builtin_amdgcn_wmma_f32_16x16x4_f32(false, a1, false, b1,
                                              (short)0, acc, false, false);
  acc = __builtin_amdgcn_wmma_f32_16x16x4_f32(false, a2, false, b2,
                                              (short)0, acc, false, false);

  // Stage D in LDS in output-memory layout: lane l holds
  // acc[v] = tile[M = v + 8*half][N = l%16]; keep only N < 12.
  __shared__ __align__(16) float stage[8][16 * T_DIM];
  float* st = stage[wave];
  if (mrow < T_DIM) {
    #pragma unroll
    for (int v = 0; v < 8; ++v)
      st[(v + 8 * half) * T_DIM + mrow] = acc[v];
  }
  __syncthreads();

  // Coalesced writeback: the LDS tile is byte-identical to the contiguous
  // 768 B destination; each lane copies float2 chunks {lane, lane+32, lane+64}
  // (ds_load_b64 / global_store_b64 with folded 0/256/512 offsets).
  float* obase = out + row_base * T_DIM;
  const int base = lane * 2;
  #pragma unroll
  for (int i = 0; i < 3; ++i) {
    const v2f val = *(const v2f*)(st + base + i * 64);
    __builtin_nontemporal_store(val, (v2f*)(obase + base + i * 64));
  }
}

// ---------------------------------------------------------------------------
extern "C" void kernel_launch(void* const* d_in, const int* in_sizes, int n_in,
                              void* d_out, int out_size, void* d_ws, size_t ws_size,
                              hipStream_t stream) {
  const float* x    = (const float*)d_in[0];
  const int*   tind = (const int*)  d_in[1];
  const float* tv1  = (const float*)d_in[2];
  const float* tv2  = (const float*)d_in[3];
  const float* nv   = (const float*)d_in[4];
  const float* kt   = (const float*)d_in[5];
  float* out  = (float*)d_out;
  float* blay = (float*)d_ws;   // 192 floats (768 B)

  adp_build_kernel<<<1, 256, 0, stream>>>(tv1, tv2, nv, kt, tind, blay);

  const long rows   = (long)in_sizes[0] / T_DIM;    // B*C*N = 8,388,608
  const long tiles  = (rows + 15) / 16;             // 524,288
  const long blocks = (tiles + 7) / 8;              // 65,536
  temporal_mix_kernel<<<(unsigned)blocks, 256, 0, stream>>>(x, blay, out, rows);
}